// Structure2VecFirstLayer_41162966565588
// MI455X (gfx1250) — compile-verified
//
#include <hip/hip_runtime.h>

typedef __attribute__((ext_vector_type(2))) float v2f;
typedef __attribute__((ext_vector_type(8))) float v8f;

#define HID    256
#define ATOM_F 256
#define BOND_F 128
#define BN_EPS 1e-5f

// ---- CDNA5 async memory->LDS copy (ASYNCcnt-tracked), inline asm ----------
// LDS destination is the byte offset in the vdst VGPR; 16 bytes per active lane.
__device__ __forceinline__ void asyncLoadB128(const void* gsrc, unsigned ldsOff) {
  asm volatile("global_load_async_to_lds_b128 %0, %1, off"
               :: "v"(ldsOff), "v"(gsrc)
               : "memory");
}
__device__ __forceinline__ void asyncWait0() {
  asm volatile("s_wait_asynccnt 0x0" ::: "memory");
}
// low 32 bits of a generic pointer to __shared__ == LDS byte offset
__device__ __forceinline__ unsigned ldsOffsetOf(const void* p) {
  return (unsigned)(unsigned long long)p;
}

// ---------------------------------------------------------------- zero
__global__ void zero_kernel(float4* __restrict__ p, int n4) {
  int i = blockIdx.x * blockDim.x + threadIdx.x;
  if (i < n4) p[i] = make_float4(0.f, 0.f, 0.f, 0.f);
}

// ---------------------------------------------------------------- edge GEMM + scatter
// h_edge[e, c] = sum_k w[e,k] * W_bond[c,k] + b_bond[c]; agg[dst[e], c] += h_edge[e, c]
// block = 512 threads = 16 waves; wave t owns output columns [16t, 16t+16).
__global__ void edge_kernel(const float* __restrict__ w,
                            const int*   __restrict__ dst,
                            const float* __restrict__ Wb,
                            const float* __restrict__ bb,
                            float*       __restrict__ agg,
                            int nEdges) {
  __shared__ float Alds[2][16 * 132];       // double-buffered 16x128 w tile (stride 132)

  const int tid  = threadIdx.x;
  const int wv   = tid >> 5;                // 0..15  (n-tile)
  const int lane = tid & 31;
  const int l15  = lane & 15;
  const int half = lane >> 4;               // 0 or 1
  const int col  = wv * 16 + l15;           // output feature column owned by this lane

  // B strip: B[k, n] = W_bond[n, k]; fragment ks covers K = 4*ks..4*ks+3.
  v2f Bfrag[32];
  {
    const float* bp = Wb + (size_t)col * BOND_F + half * 2;
#pragma unroll
    for (int ks = 0; ks < 32; ++ks)
      Bfrag[ks] = *(const v2f*)(bp + 4 * ks);
  }
  const float bias = bb[col];

  // per-lane chunk of the cooperative A-tile copy: row = wave, 4 floats per lane
  const int rr = tid >> 5;
  const int cc = (tid & 31) << 2;
  const unsigned dstOff0 = ldsOffsetOf(&Alds[0][rr * 132 + cc]);
  const unsigned dstOff1 = ldsOffsetOf(&Alds[1][rr * 132 + cc]);

  const int nTiles = nEdges >> 4;
  const int stride = gridDim.x;

  int t0 = blockIdx.x;
  if (t0 < nTiles)
    asyncLoadB128(w + (size_t)((t0 << 4) + rr) * BOND_F + cc, dstOff0);

  int cur = 0;
  for (int t = t0; t < nTiles; t += stride) {
    asyncWait0();            // my chunk of buf[cur] has landed
    __syncthreads();         // all chunks landed; buf[1-cur] free for next prefetch
    const int nt = t + stride;
    if (nt < nTiles)         // prefetch next tile, overlapped with compute below
      asyncLoadB128(w + (size_t)((nt << 4) + rr) * BOND_F + cc,
                    cur ? dstOff0 : dstOff1);

    const float* At = Alds[cur];
    v8f acc = {};
#pragma unroll
    for (int ks = 0; ks < 32; ++ks) {
      v2f a = *(const v2f*)(&At[l15 * 132 + 4 * ks + half * 2]);
      acc = __builtin_amdgcn_wmma_f32_16x16x4_f32(
          false, a, false, Bfrag[ks], (short)0, acc, false, false);
    }

    // scatter: acc[i] is element (row = half*8 + i, col)
    const int e0 = t << 4;
#pragma unroll
    for (int i = 0; i < 8; ++i) {
      const int node = dst[e0 + half * 8 + i];
      atomicAdd(agg + (size_t)node * HID + col, acc[i] + bias);
    }
    cur ^= 1;
  }
}

// ---------------------------------------------------------------- node GEMM + agg + bias + ReLU + BN partials
__global__ void node_kernel(const float* __restrict__ x,
                            const float* __restrict__ Wa,
                            const float* __restrict__ ba,
                            const float* __restrict__ agg,
                            float*       __restrict__ hOut,
                            float*       __restrict__ stats,   // [0..255]=sum  [256..511]=sumsq
                            int nNodes) {
  __shared__ float Alds[2][16 * 260];       // double-buffered 16x256 x tile (stride 260)

  const int tid  = threadIdx.x;
  const int wv   = tid >> 5;
  const int lane = tid & 31;
  const int l15  = lane & 15;
  const int half = lane >> 4;
  const int col  = wv * 16 + l15;

  v2f Bfrag[64];
  {
    const float* bp = Wa + (size_t)col * ATOM_F + half * 2;
#pragma unroll
    for (int ks = 0; ks < 64; ++ks)
      Bfrag[ks] = *(const v2f*)(bp + 4 * ks);
  }
  const float bias = ba[col];

  // per-lane chunk: row = wave, 8 floats (two b128s) per lane
  const int rr = tid >> 5;
  const int cc = (tid & 31) << 3;
  const unsigned dstOff0 = ldsOffsetOf(&Alds[0][rr * 260 + cc]);
  const unsigned dstOff1 = ldsOffsetOf(&Alds[1][rr * 260 + cc]);

  float sAcc = 0.f, qAcc = 0.f;

  const int nTiles = nNodes >> 4;
  const int stride = gridDim.x;

  int t0 = blockIdx.x;
  if (t0 < nTiles) {
    const float* src = x + (size_t)((t0 << 4) + rr) * ATOM_F + cc;
    asyncLoadB128(src,     dstOff0);
    asyncLoadB128(src + 4, dstOff0 + 16);
  }

  int cur = 0;
  for (int t = t0; t < nTiles; t += stride) {
    asyncWait0();
    __syncthreads();
    const int nt = t + stride;
    if (nt < nTiles) {
      const float* src = x + (size_t)((nt << 4) + rr) * ATOM_F + cc;
      const unsigned d = cur ? dstOff0 : dstOff1;
      asyncLoadB128(src,     d);
      asyncLoadB128(src + 4, d + 16);
    }

    const float* At = Alds[cur];
    v8f acc = {};
#pragma unroll
    for (int ks = 0; ks < 64; ++ks) {
      v2f a = *(const v2f*)(&At[l15 * 260 + 4 * ks + half * 2]);
      acc = __builtin_amdgcn_wmma_f32_16x16x4_f32(
          false, a, false, Bfrag[ks], (short)0, acc, false, false);
    }

    const int n0 = t << 4;
    float s = 0.f, q = 0.f;
#pragma unroll
    for (int i = 0; i < 8; ++i) {
      const int row = n0 + half * 8 + i;
      float v = acc[i] + bias + agg[(size_t)row * HID + col];
      v = fmaxf(v, 0.f);
      hOut[(size_t)row * HID + col] = v;
      s += v;
      q += v * v;
    }
    // combine the two half-wave partials for this column (lanes L and L+16 share col)
    s += __shfl_down(s, 16, 32);
    q += __shfl_down(q, 16, 32);
    if (half == 0) { sAcc += s; qAcc += q; }
    cur ^= 1;
  }

  if (half == 0) {
    atomicAdd(stats + col,       sAcc);
    atomicAdd(stats + HID + col, qAcc);
  }
}

// ---------------------------------------------------------------- BN stats -> scale/shift
__global__ void bn_stats_kernel(float* __restrict__ stats,
                                const float* __restrict__ gamma,
                                const float* __restrict__ beta,
                                int nNodes) {
  const int f = threadIdx.x;                // 256 threads
  const float invN = 1.f / (float)nNodes;
  const float mean = stats[f] * invN;
  const float var  = stats[HID + f] * invN - mean * mean;
  const float inv  = rsqrtf(var + BN_EPS);
  const float sc   = gamma[f] * inv;
  stats[2 * HID + f] = sc;                  // scale
  stats[3 * HID + f] = beta[f] - mean * sc; // shift
}

// ---------------------------------------------------------------- apply BN in place
__global__ void bn_apply_kernel(float4* __restrict__ out,
                                const float* __restrict__ stats,
                                int n4) {
  int i = blockIdx.x * blockDim.x + threadIdx.x;
  if (i >= n4) return;
  const int c = (i * 4) & (HID - 1);
  const float* scale = stats + 2 * HID;
  const float* shift = stats + 3 * HID;
  float4 v = out[i];
  v.x = v.x * scale[c]     + shift[c];
  v.y = v.y * scale[c + 1] + shift[c + 1];
  v.z = v.z * scale[c + 2] + shift[c + 2];
  v.w = v.w * scale[c + 3] + shift[c + 3];
  out[i] = v;
}

// ---------------------------------------------------------------- launcher
extern "C" void kernel_launch(void* const* d_in, const int* in_sizes, int n_in,
                              void* d_out, int out_size, void* d_ws, size_t ws_size,
                              hipStream_t stream) {
  const float* x      = (const float*)d_in[0];
  const float* w      = (const float*)d_in[1];
  const int*   dst    = (const int*)  d_in[2];
  const float* W_atom = (const float*)d_in[3];
  const float* b_atom = (const float*)d_in[4];
  const float* W_bond = (const float*)d_in[5];
  const float* b_bond = (const float*)d_in[6];
  const float* gamma  = (const float*)d_in[7];
  const float* beta   = (const float*)d_in[8];
  float* out = (float*)d_out;

  const int nNodes = in_sizes[0] / ATOM_F;   // 50000
  const int nEdges = in_sizes[1] / BOND_F;   // 800000

  float* agg   = (float*)d_ws;                       // [nNodes * HID]
  float* stats = agg + (size_t)nNodes * HID;         // [4 * 256] sum/sumsq/scale/shift

  // 1) zero agg + stats
  const int n4 = (nNodes * HID + 4 * HID) / 4;
  zero_kernel<<<(n4 + 255) / 256, 256, 0, stream>>>((float4*)d_ws, n4);

  // 2) edge GEMM + scatter-add
  const int eTiles  = nEdges / 16;
  const int eBlocks = eTiles < 2000 ? eTiles : 2000;
  edge_kernel<<<eBlocks, 512, 0, stream>>>(w, dst, W_bond, b_bond, agg, nEdges);

  // 3) node GEMM + agg + bias + ReLU + BN partial sums
  const int nTiles  = nNodes / 16;
  const int nBlocks = nTiles < 1024 ? nTiles : 1024;
  node_kernel<<<nBlocks, 512, 0, stream>>>(x, W_atom, b_atom, agg, out, stats, nNodes);

  // 4) finalize BN stats
  bn_stats_kernel<<<1, 256, 0, stream>>>(stats, gamma, beta, nNodes);

  // 5) apply BN in place on d_out
  const int o4 = (nNodes * HID) / 4;
  bn_apply_kernel<<<(o4 + 255) / 256, 256, 0, stream>>>((float4*)out, stats, o4);
}